// GCNLayer_45105746542853
// MI455X (gfx1250) — compile-verified
//
#include <hip/hip_runtime.h>
#include <hip/hip_bf16.h>

#define N_NODES 50000
#define N_EDGES 800000
#define D_IN    256
#define D_OUT   64

// Padded pair-row stride (in float2 units): 80*2 banks = 160 ≡ 32 (mod 64),
// so half-wave 0 (banks 0..31) and half-wave 1 (banks 32..63) never collide.
#define WSTRIDE 80

typedef float v2f __attribute__((ext_vector_type(2)));
typedef float v8f __attribute__((ext_vector_type(8)));

// ---------------------------------------------------------------------------
// Kernel 1: xw = x @ W   (fp32 WMMA 16x16x4)
// Block = 256 threads = 8 waves. Each wave computes a 16x64 strip of xw.
// Weight staged in LDS K-pair interleaved: wlds[k/2][n] = {W[k][n], W[k+1][n]}
// so each B operand is one aligned ds_load_b64 into an even VGPR pair.
// ---------------------------------------------------------------------------
__global__ __launch_bounds__(256)
void gcn_gemm_wmma(const float* __restrict__ x,
                   const float* __restrict__ w,
                   float* __restrict__ xw)
{
    __shared__ v2f wlds[(D_IN / 2) * WSTRIDE];   // 128*80*8 = 80 KB

    // Cooperative interleaving load of the whole weight matrix into LDS
    {
        const int npair = (D_IN / 2) * D_OUT;    // 8192 float2 elements
        for (int i = threadIdx.x; i < npair; i += 256) {
            const int k2 = i >> 6;               // K pair index 0..127
            const int n  = i & 63;               // column 0..63
            v2f p;
            p.x = w[(2 * k2 + 0) * D_OUT + n];
            p.y = w[(2 * k2 + 1) * D_OUT + n];
            wlds[k2 * WSTRIDE + n] = p;
        }
    }
    __syncthreads();

    const int lane = threadIdx.x & 31;
    const int wave = threadIdx.x >> 5;
    const int half = lane >> 4;       // which half-wave (K split)
    const int m    = lane & 15;       // row within tile (A) / col within tile (B,C)

    const int rowBase = blockIdx.x * 128 + wave * 16;   // 8 waves * 16 rows
    // Clamp A-row for loads so EXEC stays all-ones for WMMA (stores predicated)
    int arow = rowBase + m;
    if (arow > N_NODES - 1) arow = N_NODES - 1;
    const float* __restrict__ xrow = x + (long)arow * D_IN;

    v8f acc0 = {}, acc1 = {}, acc2 = {}, acc3 = {};

    #pragma unroll 4
    for (int k0 = 0; k0 < D_IN; k0 += 4) {
        // This half-wave supplies K = {k0+2*half, k0+2*half+1}
        // A operand: row m, one 8-byte global load
        v2f a = *reinterpret_cast<const v2f*>(xrow + k0 + 2 * half);

        // B operands: one aligned b64 LDS load per column tile
        const v2f* __restrict__ brow = wlds + ((k0 >> 1) + half) * WSTRIDE + m;
        v2f b0 = brow[ 0];
        v2f b1 = brow[16];
        v2f b2 = brow[32];
        v2f b3 = brow[48];

        acc0 = __builtin_amdgcn_wmma_f32_16x16x4_f32(false, a, false, b0, (short)0, acc0, false, false);
        acc1 = __builtin_amdgcn_wmma_f32_16x16x4_f32(false, a, false, b1, (short)0, acc1, false, false);
        acc2 = __builtin_amdgcn_wmma_f32_16x16x4_f32(false, a, false, b2, (short)0, acc2, false, false);
        acc3 = __builtin_amdgcn_wmma_f32_16x16x4_f32(false, a, false, b3, (short)0, acc3, false, false);
    }

    // Store D: VGPR r holds M = r + 8*half, N = colTile*16 + m
    #pragma unroll
    for (int r = 0; r < 8; ++r) {
        const int orow = rowBase + r + 8 * half;
        if (orow < N_NODES) {
            float* __restrict__ o = xw + (long)orow * D_OUT;
            o[ 0 + m] = acc0[r];
            o[16 + m] = acc1[r];
            o[32 + m] = acc2[r];
            o[48 + m] = acc3[r];
        }
    }
}

// ---------------------------------------------------------------------------
// Kernel 2: out[n][c] = bias[c]  (init accumulation buffer; d_out is poisoned)
// ---------------------------------------------------------------------------
__global__ __launch_bounds__(256)
void gcn_init_bias(float* __restrict__ out, const float* __restrict__ bias)
{
    int i = blockIdx.x * 256 + threadIdx.x;
    if (i < N_NODES * D_OUT) {
        out[i] = bias[i & (D_OUT - 1)];
    }
}

// ---------------------------------------------------------------------------
// Kernel 3: COO SpMM scatter: out[dst] += val * xw[src]
// 16 lanes per edge, float4 gather, 4 hardware fp32 atomics per lane.
// xw (12.8 MB) and out (12.8 MB) are both L2-resident (192 MB L2).
// ---------------------------------------------------------------------------
__global__ __launch_bounds__(256)
void gcn_spmm_scatter(const float* __restrict__ xw,
                      const int* __restrict__ esrc,
                      const int* __restrict__ edst,
                      const float* __restrict__ evals,
                      float* __restrict__ out)
{
    int idx = blockIdx.x * 256 + threadIdx.x;
    int e   = idx >> 4;              // 16 threads per edge
    if (e >= N_EDGES) return;
    int q   = idx & 15;              // which float4 chunk of the 64 columns

    const float v = evals[e];
    const int  s  = esrc[e];
    const int  d  = edst[e];

    const float4 g = reinterpret_cast<const float4*>(xw + (long)s * D_OUT)[q];
    float* __restrict__ o = out + (long)d * D_OUT + q * 4;

    unsafeAtomicAdd(o + 0, v * g.x);
    unsafeAtomicAdd(o + 1, v * g.y);
    unsafeAtomicAdd(o + 2, v * g.z);
    unsafeAtomicAdd(o + 3, v * g.w);
}

// ---------------------------------------------------------------------------
// Kernel 4: in-place ReLU
// ---------------------------------------------------------------------------
__global__ __launch_bounds__(256)
void gcn_relu(float* __restrict__ out)
{
    int i = blockIdx.x * 256 + threadIdx.x;
    if (i < N_NODES * D_OUT) {
        float t = out[i];
        out[i] = t > 0.0f ? t : 0.0f;
    }
}

// ---------------------------------------------------------------------------
extern "C" void kernel_launch(void* const* d_in, const int* in_sizes, int n_in,
                              void* d_out, int out_size, void* d_ws, size_t ws_size,
                              hipStream_t stream)
{
    const float* x     = (const float*)d_in[0];   // [N_NODES, D_IN]
    const int*   esrc  = (const int*)  d_in[1];   // [N_EDGES]
    const int*   edst  = (const int*)  d_in[2];   // [N_EDGES]
    const float* evals = (const float*)d_in[3];   // [N_EDGES]
    const float* w     = (const float*)d_in[4];   // [D_IN, D_OUT]
    const float* bias  = (const float*)d_in[5];   // [D_OUT]
    float*       out   = (float*)d_out;           // [N_NODES, D_OUT]
    float*       xw    = (float*)d_ws;            // scratch: N_NODES*D_OUT*4 = 12.8 MB

    (void)in_sizes; (void)n_in; (void)out_size; (void)ws_size;

    // 1) Dense GEMM via fp32 WMMA: 128 rows per block
    {
        dim3 grid((N_NODES + 127) / 128);
        gcn_gemm_wmma<<<grid, 256, 0, stream>>>(x, w, xw);
    }
    // 2) Init output with bias
    {
        dim3 grid((N_NODES * D_OUT + 255) / 256);
        gcn_init_bias<<<grid, 256, 0, stream>>>(out, bias);
    }
    // 3) Edge scatter with hardware fp32 atomics
    {
        long threads = (long)N_EDGES * 16;
        dim3 grid((unsigned)((threads + 255) / 256));
        gcn_spmm_scatter<<<grid, 256, 0, stream>>>(xw, esrc, edst, evals, out);
    }
    // 4) ReLU
    {
        dim3 grid((N_NODES * D_OUT + 255) / 256);
        gcn_relu<<<grid, 256, 0, stream>>>(out);
    }
}